// FusedTensor_4277787427181
// MI455X (gfx1250) — compile-verified
//
#include <hip/hip_runtime.h>
#include <cmath>
#include <complex>

// ---------------------------------------------------------------------------
// Problem constants (reference: MAX_DEGREE=2, LTILDE=1, b=16384, f=128)
// ---------------------------------------------------------------------------
constexpr int FDIM   = 128;            // feature dim
constexpr int NC     = 9;              // (L+1)^2 coefficients
constexpr int ROWS   = 8;              // batch rows per block
constexpr int BLK    = 128;            // one thread per feature
constexpr int ROW_F  = 2 * NC * FDIM;  // floats per input row  (2304)
constexpr int CH_IN  = ROW_F / 4;      // float4 chunks per input row (576)
constexpr int CHUNKS = 2 * CH_IN;      // chunks per (in1,in2) row pair (1152)
constexpr int SBUF_F = 2 * ROW_F;      // floats per LDS buffer (4608)

constexpr int DEGn[NC] = {0, 1,1,1, 2,2,2,2,2};

// Structural nonzeros of CG[i][j][n] for 1x1 -> (0,1,2) in real SH basis
// (axis order y,z,x): l=0 diag, l=1 cross-product, l=2 sym-traceless.
constexpr int NNZ = 20;
constexpr int NZI[NNZ] = {0,1,2, 1,2, 0,2, 0,1, 0,2, 0,1, 0,1,2, 1,2, 0,2};
constexpr int NZJ[NNZ] = {0,1,2, 2,1, 2,0, 1,0, 2,0, 1,0, 0,1,2, 2,1, 0,2};
constexpr int NZN[NNZ] = {0,0,0, 1,1, 2,2, 3,3, 4,4, 5,5, 6,6,6, 7,7, 8,8};

struct CGc { float v[81]; };   // [i][j][n] -> v[(i*3+j)*9+n], kernarg by value

typedef int v4i __attribute__((ext_vector_type(4)));

// ---------------------------------------------------------------------------
// CDNA5 async global->LDS path (guarded; falls back to inline asm, then to
// plain copies on the host parse).
// ---------------------------------------------------------------------------
#if defined(__AMDGCN__)
#define CGK_DEV 1
#if __has_builtin(__builtin_amdgcn_global_load_async_to_lds_b128)
#define CGK_ASYNC_BUILTIN 1
#else
#define CGK_ASYNC_BUILTIN 0
#endif
#if __has_builtin(__builtin_amdgcn_s_wait_asynccnt)
#define CGK_WAIT_BUILTIN 1
#else
#define CGK_WAIT_BUILTIN 0
#endif
#else
#define CGK_DEV 0
#endif

__device__ __forceinline__ void async_cp16(void* lds_dst, const void* gsrc) {
#if CGK_DEV
#if CGK_ASYNC_BUILTIN
  // Probe-confirmed signature: (v4i AS1*, v4i AS3*, imm offset, imm cpol)
  __builtin_amdgcn_global_load_async_to_lds_b128(
      (__attribute__((address_space(1))) v4i*)(void*)gsrc,
      (__attribute__((address_space(3))) v4i*)lds_dst,
      0, 0);
#else
  unsigned lo = (unsigned)(unsigned long long)
      (__attribute__((address_space(3))) v4i*)lds_dst;
  asm volatile("global_load_async_to_lds_b128 %0, %1, off"
               :: "v"(lo), "v"(gsrc) : "memory");
#endif
#else
  __builtin_memcpy(lds_dst, gsrc, 16);
#endif
}

template <int N>
__device__ __forceinline__ void wait_async() {
#if CGK_DEV
#if CGK_WAIT_BUILTIN
  __builtin_amdgcn_s_wait_asynccnt(N);
#else
  asm volatile("s_wait_asynccnt %0" :: "n"(N) : "memory");
#endif
#endif
}

// ---------------------------------------------------------------------------
// Small dense/sparse tensor helpers (fully unrolled, constant indices)
// ---------------------------------------------------------------------------
template <bool SP>
__device__ __forceinline__ void enc3(const CGc& cg, const float t[NC], float A[3][3]) {
#pragma unroll
  for (int i = 0; i < 3; ++i)
#pragma unroll
    for (int j = 0; j < 3; ++j) A[i][j] = 0.0f;
  if (SP) {
#pragma unroll
    for (int k = 0; k < NNZ; ++k)
      A[NZI[k]][NZJ[k]] = fmaf(cg.v[(NZI[k]*3 + NZJ[k])*9 + NZN[k]],
                               t[NZN[k]], A[NZI[k]][NZJ[k]]);
  } else {
#pragma unroll
    for (int i = 0; i < 3; ++i)
#pragma unroll
      for (int j = 0; j < 3; ++j)
#pragma unroll
        for (int n = 0; n < NC; ++n)
          A[i][j] = fmaf(cg.v[(i*3 + j)*9 + n], t[n], A[i][j]);
  }
}

template <bool SP>
__device__ __forceinline__ void dec3(const CGc& cg, const float Z[3][3], float d[NC]) {
#pragma unroll
  for (int n = 0; n < NC; ++n) d[n] = 0.0f;
  if (SP) {
#pragma unroll
    for (int k = 0; k < NNZ; ++k)
      d[NZN[k]] = fmaf(cg.v[(NZI[k]*3 + NZJ[k])*9 + NZN[k]],
                       Z[NZI[k]][NZJ[k]], d[NZN[k]]);
  } else {
#pragma unroll
    for (int n = 0; n < NC; ++n)
#pragma unroll
      for (int i = 0; i < 3; ++i)
#pragma unroll
        for (int j = 0; j < 3; ++j)
          d[n] = fmaf(cg.v[(i*3 + j)*9 + n], Z[i][j], d[n]);
  }
}

__device__ __forceinline__ void mm3(const float A[3][3], const float B[3][3], float C[3][3]) {
#pragma unroll
  for (int i = 0; i < 3; ++i)
#pragma unroll
    for (int j = 0; j < 3; ++j) {
      float acc = A[i][0] * B[0][j];
      acc = fmaf(A[i][1], B[1][j], acc);
      acc = fmaf(A[i][2], B[2][j], acc);
      C[i][j] = acc;
    }
}

// ---------------------------------------------------------------------------
// Kernel: one thread per feature column, ROWS batch rows per block,
// double-buffered async global->LDS staging of (inputs1,inputs2) rows.
// ---------------------------------------------------------------------------
template <bool SP>
__global__ void __launch_bounds__(BLK)
sph_tensor_kernel(const float* __restrict__ in1, const float* __restrict__ in2,
                  const float* __restrict__ w1,  const float* __restrict__ w2,
                  const float* __restrict__ w3,  float* __restrict__ out,
                  CGc cg, int B) {
  __shared__ float sbuf[2][SBUF_F];

  const int f       = threadIdx.x;
  const int rowBase = blockIdx.x * ROWS;

  // Per-feature weights (reused across all rows of this block)
  float W1[2][3], W2[2][3], W3[4][3];
#pragma unroll
  for (int p = 0; p < 2; ++p)
#pragma unroll
    for (int l = 0; l < 3; ++l) {
      W1[p][l] = w1[(p*3 + l)*FDIM + f];
      W2[p][l] = w2[(p*3 + l)*FDIM + f];
    }
#pragma unroll
  for (int p = 0; p < 4; ++p)
#pragma unroll
    for (int l = 0; l < 3; ++l)
      W3[p][l] = w3[(p*3 + l)*FDIM + f];

  auto stage = [&](int buf, int row) {
    const float* g1 = in1 + (size_t)row * ROW_F;
    const float* g2 = in2 + (size_t)row * ROW_F;
#pragma unroll
    for (int k = 0; k < CHUNKS / BLK; ++k) {   // 9 b128 async ops per thread
      const int c = f + k * BLK;
      const float* g = (c < CH_IN) ? (g1 + c * 4) : (g2 + (c - CH_IN) * 4);
      async_cp16(&sbuf[buf][c * 4], g);
    }
  };

  if (rowBase < B) stage(0, rowBase);

#pragma unroll 1
  for (int r = 0; r < ROWS; ++r) {
    const int  row = rowBase + r;
    const bool cur = row < B;
    const bool nxt = (r + 1 < ROWS) && (row + 1 < B);
    if (nxt) { stage((r + 1) & 1, row + 1); wait_async<CHUNKS / BLK>(); }
    else     { wait_async<0>(); }
    __syncthreads();

    if (cur) {
      const float* s = sbuf[r & 1];

      // t = parity-masked weight contraction (one product per coefficient)
      float t1p[NC], t1m[NC], t2p[NC], t2m[NC];
#pragma unroll
      for (int n = 0; n < NC; ++n) {
        const int d  = DEGn[n];
        const int pe = d & 1, po = pe ^ 1;
        const float xe = s[(pe*NC + n)*FDIM + f];
        const float xo = s[(po*NC + n)*FDIM + f];
        const float ye = s[ROW_F + (pe*NC + n)*FDIM + f];
        const float yo = s[ROW_F + (po*NC + n)*FDIM + f];
        t1p[n] = xe * W1[pe][d];
        t1m[n] = xo * W1[po][d];
        t2p[n] = ye * W2[pe][d];
        t2m[n] = yo * W2[po][d];
      }

      float Xp[3][3], Xm[3][3], Yp[3][3], Ym[3][3], Z[3][3], dt[NC], ze[NC], zo[NC];
      enc3<SP>(cg, t1p, Xp);
      enc3<SP>(cg, t1m, Xm);
      enc3<SP>(cg, t2p, Yp);
      enc3<SP>(cg, t2m, Ym);

      mm3(Xp, Yp, Z); dec3<SP>(cg, Z, dt);
#pragma unroll
      for (int n = 0; n < NC; ++n) ze[n] = dt[n] * W3[0][DEGn[n]];
      mm3(Xm, Ym, Z); dec3<SP>(cg, Z, dt);
#pragma unroll
      for (int n = 0; n < NC; ++n) ze[n] = fmaf(dt[n], W3[1][DEGn[n]], ze[n]);
      mm3(Xp, Ym, Z); dec3<SP>(cg, Z, dt);
#pragma unroll
      for (int n = 0; n < NC; ++n) zo[n] = dt[n] * W3[2][DEGn[n]];
      mm3(Xm, Yp, Z); dec3<SP>(cg, Z, dt);
#pragma unroll
      for (int n = 0; n < NC; ++n) zo[n] = fmaf(dt[n], W3[3][DEGn[n]], zo[n]);

      float* o = out + (size_t)row * (2 * NC * FDIM);
#pragma unroll
      for (int n = 0; n < NC; ++n) {
        const bool ev = (DEGn[n] & 1) == 0;
        o[n*FDIM + f]            = ev ? ze[n] : zo[n];
        o[NC*FDIM + n*FDIM + f]  = ev ? zo[n] : ze[n];
      }
    }
    __syncthreads();   // protect buffer (r&1) before it is restaged
  }
}

// ---------------------------------------------------------------------------
// Host: exact port of the reference CG construction (Racah + real-SH basis)
// ---------------------------------------------------------------------------
static double factd(int n) { double r = 1.0; for (int i = 2; i <= n; ++i) r *= (double)i; return r; }

static double cg_complex_h(int j1, int j2, int j3, int m1, int m2, int m3) {
  if (m1 + m2 != m3) return 0.0;
  if (j3 < std::abs(j1 - j2) || j3 > j1 + j2) return 0.0;
  double pre = std::sqrt((double)(2*j3 + 1) * factd(j3 + j1 - j2) * factd(j3 - j1 + j2) *
                         factd(j1 + j2 - j3) / factd(j1 + j2 + j3 + 1));
  pre *= std::sqrt(factd(j3 + m3) * factd(j3 - m3) * factd(j1 - m1) * factd(j1 + m1) *
                   factd(j2 - m2) * factd(j2 + m2));
  double s = 0.0;
  for (int k = 0; k <= j1 + j2; ++k) {
    const int d0 = k, d1 = j1 + j2 - j3 - k, d2 = j1 - m1 - k,
              d3 = j2 + m2 - k, d4 = j3 - j2 + m1 + k, d5 = j3 - j1 - m2 + k;
    if (d0 < 0 || d1 < 0 || d2 < 0 || d3 < 0 || d4 < 0 || d5 < 0) continue;
    const double denom = factd(d0)*factd(d1)*factd(d2)*factd(d3)*factd(d4)*factd(d5);
    s += ((k & 1) ? -1.0 : 1.0) / denom;
  }
  return pre * s;
}

static void umat_h(int l, std::complex<double>* u) {  // (2l+1)x(2l+1) row-major
  const int m = 2*l + 1;
  for (int i = 0; i < m*m; ++i) u[i] = 0.0;
  u[l*m + l] = 1.0;
  const double is2 = 1.0 / std::sqrt(2.0);
  for (int mm = 1; mm <= l; ++mm) {
    const double sgn = (mm & 1) ? -1.0 : 1.0;
    u[(l + mm)*m + (l + mm)] = sgn * is2;
    u[(l + mm)*m + (l - mm)] = is2;
    u[(l - mm)*m + (l - mm)] = std::complex<double>(0.0, is2);
    u[(l - mm)*m + (l + mm)] = std::complex<double>(0.0, -sgn * is2);
  }
}

static void build_cg_h(float* out81) {
  std::complex<double> u1[9];
  umat_h(1, u1);
  for (int i = 0; i < 81; ++i) out81[i] = 0.0f;
  for (int l3 = 0; l3 <= 2; ++l3) {
    const int md = 2*l3 + 1;
    std::complex<double> u3[25];
    umat_h(l3, u3);
    double cgc[3][3][5] = {};
    for (int m1 = -1; m1 <= 1; ++m1)
      for (int m2 = -1; m2 <= 1; ++m2) {
        const int m3 = m1 + m2;
        if (m3 >= -l3 && m3 <= l3)
          cgc[1 + m1][1 + m2][l3 + m3] = cg_complex_h(1, 1, l3, m1, m2, m3);
      }
    for (int a = 0; a < 3; ++a)
      for (int b = 0; b < 3; ++b)
        for (int c = 0; c < md; ++c) {
          std::complex<double> acc(0.0, 0.0);
          for (int p = 0; p < 3; ++p)
            for (int q = 0; q < 3; ++q)
              for (int rr = 0; rr < md; ++rr)
                acc += u1[a*3 + p] * u1[b*3 + q] * std::conj(u3[c*md + rr]) * cgc[p][q][rr];
          const double val = ((2 + l3) % 2 == 0) ? acc.real() : acc.imag();
          out81[(a*3 + b)*9 + l3*l3 + c] = (float)val;
        }
  }
}

// ---------------------------------------------------------------------------
extern "C" void kernel_launch(void* const* d_in, const int* in_sizes, int n_in,
                              void* d_out, int out_size, void* d_ws, size_t ws_size,
                              hipStream_t stream) {
  (void)n_in; (void)out_size; (void)d_ws; (void)ws_size;
  const float* in1 = (const float*)d_in[0];
  const float* in2 = (const float*)d_in[1];
  const float* w1  = (const float*)d_in[2];
  const float* w2  = (const float*)d_in[3];
  const float* w3  = (const float*)d_in[4];
  float* out = (float*)d_out;

  const int B = in_sizes[0] / ROW_F;

  CGc cg;
  build_cg_h(cg.v);

  // Verify assumed sparsity pattern; fall back to dense contraction otherwise.
  bool allowed[81] = {};
  for (int k = 0; k < NNZ; ++k) allowed[(NZI[k]*3 + NZJ[k])*9 + NZN[k]] = true;
  bool sparse_ok = true;
  for (int i = 0; i < 81; ++i)
    if (!allowed[i] && std::fabs(cg.v[i]) > 1e-6f) { sparse_ok = false; break; }

  dim3 grid((B + ROWS - 1) / ROWS), block(BLK);
  if (sparse_ok)
    sph_tensor_kernel<true><<<grid, block, 0, stream>>>(in1, in2, w1, w2, w3, out, cg, B);
  else
    sph_tensor_kernel<false><<<grid, block, 0, stream>>>(in1, in2, w1, w2, w3, out, cg, B);
}